// YoloHeadsLoss_64252710748541
// MI455X (gfx1250) — compile-verified
//
#include <hip/hip_runtime.h>
#include <hip/hip_bf16.h>
#include <stdint.h>

#define B_    32
#define L_    8400
#define N_    100
#define K3_   204   /* K=68 keypoints * 3 */
#define TOPK_ 13
#define TPB   256
#define VPT   33    /* ceil(8400/256) */
#define EPS_  1e-9f

typedef float    v4f __attribute__((ext_vector_type(4)));
typedef float    v2f __attribute__((ext_vector_type(2)));
typedef unsigned v4u __attribute__((ext_vector_type(4)));

/* ---------------- workspace layout (32-bit words) ---------------- */
constexpr size_t WS_BITS   = 0;                               /* B*L*4 u32 topk bitmask */
constexpr size_t WS_MAXAM  = WS_BITS  + (size_t)B_*L_*4;      /* B*N f32 (as u32 bits)  */
constexpr size_t WS_MAXIOU = WS_MAXAM + (size_t)B_*N_;
constexpr size_t WS_AMVAL  = WS_MAXIOU+ (size_t)B_*N_;        /* B*L f32 */
constexpr size_t WS_GIDX   = WS_AMVAL + (size_t)B_*L_;        /* B*L u32 */
constexpr size_t WS_POS    = WS_GIDX  + (size_t)B_*L_;        /* B*L u32 */
constexpr size_t WS_ZERO_WORDS = WS_AMVAL;                    /* bits + maxam + maxiou (div by 4) */

/* ---------------- output layout (floats, tuple order) ---------------- */
constexpr size_t O_LAB   = 0;
constexpr size_t O_BBOX  = O_LAB  + (size_t)B_*L_;
constexpr size_t O_SCORE = O_BBOX + (size_t)B_*L_*4;
constexpr size_t O_POSE  = O_SCORE+ (size_t)B_*L_;            /* 16B aligned */
constexpr size_t O_VERT  = O_POSE + (size_t)B_*L_*K3_;        /* 16B aligned */
constexpr size_t O_ROT   = O_VERT + (size_t)B_*L_*K3_;
constexpr size_t O_GID   = O_ROT  + (size_t)B_*L_*9;

__device__ __forceinline__ float iou_box(float gx1,float gy1,float gx2,float gy2,
                                         float px1,float py1,float px2,float py2){
  float ltx=fmaxf(gx1,px1), lty=fmaxf(gy1,py1);
  float rbx=fminf(gx2,px2), rby=fminf(gy2,py2);
  float w=fmaxf(rbx-ltx,0.f), h=fmaxf(rby-lty,0.f);
  float inter=w*h;
  float ga=(gx2-gx1)*(gy2-gy1), pa=(px2-px1)*(py2-py1);
  return inter/(ga+pa-inter+EPS_);
}

__global__ void init_kernel(v4u* __restrict__ ws4, size_t nvec){
  size_t i = (size_t)blockIdx.x*blockDim.x + threadIdx.x;
  if (i < nvec) ws4[i] = (v4u){0u,0u,0u,0u};
}

/* ---------- pass 1: per-(b,n) top-13 anchors -> scatter bit n into mask ---------- */
__global__ __launch_bounds__(TPB)
void topk_kernel(const float* __restrict__ pred_scores, const float* __restrict__ pred_bboxes,
                 const float* __restrict__ anchor_points, const float* __restrict__ gt_bboxes,
                 const float* __restrict__ pad_gt_mask, unsigned* __restrict__ ws)
{
  const int b = blockIdx.y, n = blockIdx.x;
  __shared__ float g[4];
  __shared__ float spad;
  __shared__ float sval[TPB];
  __shared__ int   sidx[TPB];
  __shared__ int   best_l[TOPK_];
  if (threadIdx.x < 4) g[threadIdx.x] = gt_bboxes[((size_t)b*N_+n)*4 + threadIdx.x];
  if (threadIdx.x == 4) spad = pad_gt_mask[(size_t)b*N_+n];
  __syncthreads();
  const float gx1=g[0], gy1=g[1], gx2=g[2], gy2=g[3];

  float vals[VPT];
  #pragma unroll
  for (int j=0;j<VPT;j++){
    int l = threadIdx.x + j*TPB;
    float m = -1.0f;
    if (l < L_){
      v4f p = *(const v4f*)(pred_bboxes + ((size_t)b*L_ + l)*4);
      float iou = iou_box(gx1,gy1,gx2,gy2,p.x,p.y,p.z,p.w);
      float s   = pred_scores[(size_t)b*L_ + l];
      float i2  = iou*iou;
      float al  = s*i2*i2*i2;                              /* score^1 * iou^6 */
      v2f  ap   = *(const v2f*)(anchor_points + (size_t)l*2);
      float dmin= fminf(fminf(ap.x-gx1, ap.y-gy1), fminf(gx2-ap.x, gy2-ap.y));
      m = (dmin > EPS_) ? al : 0.0f;                       /* alignment * is_in_gts */
    }
    vals[j] = m;
  }

  for (int t=0;t<TOPK_;t++){
    float bv = -2.0f; int bj = 0;
    #pragma unroll
    for (int j=0;j<VPT;j++) if (vals[j] > bv){ bv = vals[j]; bj = j; }
    sval[threadIdx.x] = bv;
    sidx[threadIdx.x] = bj*TPB + threadIdx.x;              /* global anchor index */
    __syncthreads();
    for (int s=TPB/2; s>0; s>>=1){
      if (threadIdx.x < s){
        float ov = sval[threadIdx.x+s]; int oi = sidx[threadIdx.x+s];
        if (ov > sval[threadIdx.x] || (ov == sval[threadIdx.x] && oi < sidx[threadIdx.x])){
          sval[threadIdx.x]=ov; sidx[threadIdx.x]=oi;
        }
      }
      __syncthreads();
    }
    int wl = sidx[0];
    if (threadIdx.x == 0) best_l[t] = wl;
    if ((wl & (TPB-1)) == (int)threadIdx.x){               /* owner clears winner */
      int jj = wl / TPB;
      #pragma unroll
      for (int q=0;q<VPT;q++) if (q==jj) vals[q] = -2.0f;
    }
    __syncthreads();
  }

  if (spad > 0.f && threadIdx.x < TOPK_){                  /* pad<=0 -> dup@0 -> zeroed */
    int l = best_l[threadIdx.x];
    atomicOr(&ws[WS_BITS + ((size_t)b*L_ + l)*4 + (n>>5)], 1u << (n&31));
  }
}

/* ---------- pass 2: per-anchor reduction over N ---------- */
__global__ __launch_bounds__(TPB)
void assign_kernel(const float* __restrict__ pred_scores, const float* __restrict__ pred_bboxes,
                   const float* __restrict__ anchor_points, const float* __restrict__ gt_bboxes,
                   const float* __restrict__ pad_gt_mask, unsigned* __restrict__ ws)
{
  __shared__ float sg[N_*4];
  __shared__ float sp[N_];
  const int b = blockIdx.y;

  /* CDNA5 async DMA: stage this batch's 100 GT boxes (1600B) global -> LDS */
  if (threadIdx.x < 32){
    unsigned long long base = (unsigned long long)(gt_bboxes + (size_t)b*N_*4);
    for (int r = threadIdx.x; r < N_; r += 32){
      unsigned lds_off = (unsigned)(size_t)(&sg[r*4]);     /* LDS byte offset */
      unsigned goff    = (unsigned)(r*16);
      asm volatile("global_load_async_to_lds_b128 %0, %1, %2"
                   :: "v"(lds_off), "v"(goff), "s"(base) : "memory");
    }
    asm volatile("s_wait_asynccnt 0" ::: "memory");
  }
  if (threadIdx.x < N_) sp[threadIdx.x] = pad_gt_mask[(size_t)b*N_ + threadIdx.x];
  __syncthreads();

  const int l = blockIdx.x*TPB + threadIdx.x;
  if (l >= L_) return;

  v4f p = *(const v4f*)(pred_bboxes + ((size_t)b*L_ + l)*4);
  const float px1=p.x, py1=p.y, px2=p.z, py2=p.w;
  v2f ap = *(const v2f*)(anchor_points + (size_t)l*2);
  const float ax = ap.x, ay = ap.y;
  const float sc = pred_scores[(size_t)b*L_ + l];
  v4u bwv = *(const v4u*)(ws + WS_BITS + ((size_t)b*L_ + l)*4);
  const unsigned w0=bwv.x, w1=bwv.y, w2=bwv.z, w3=bwv.w;

  int cnt=0, firstn=-1, bioun=0;
  float biou=-1.0f;
  for (int n=0;n<N_;n++){
    float gx1=sg[n*4], gy1=sg[n*4+1], gx2=sg[n*4+2], gy2=sg[n*4+3];
    float iou = iou_box(gx1,gy1,gx2,gy2,px1,py1,px2,py2);
    if (iou > biou){ biou=iou; bioun=n; }                  /* first-max tie-break */
    unsigned word = (n<32)?w0:((n<64)?w1:((n<96)?w2:w3));
    unsigned bit  = (word >> (n&31)) & 1u;
    float dmin = fminf(fminf(ax-gx1, ay-gy1), fminf(gx2-ax, gy2-ay));
    if (bit && (dmin > EPS_) && sp[n] > 0.f){ cnt++; if (firstn < 0) firstn = n; }
  }

  int nstar, pos;
  if (cnt == 0){ pos=0; nstar=0; }
  else if (cnt == 1){ pos=1; nstar=firstn; }
  else { pos=1; nstar=bioun; }                             /* multi-GT -> max-IoU one-hot */

  float amv = 0.f;
  if (pos){
    float gx1=sg[nstar*4], gy1=sg[nstar*4+1], gx2=sg[nstar*4+2], gy2=sg[nstar*4+3];
    float iou = iou_box(gx1,gy1,gx2,gy2,px1,py1,px2,py2);
    float i2=iou*iou;
    amv = sc*i2*i2*i2;                                     /* am = alignment * mask */
    atomicMax(&ws[WS_MAXAM  + (size_t)b*N_ + nstar], __float_as_uint(amv));
    atomicMax(&ws[WS_MAXIOU + (size_t)b*N_ + nstar], __float_as_uint(iou));
  }
  ws[WS_GIDX + (size_t)b*L_ + l] = (unsigned)nstar;
  ws[WS_POS  + (size_t)b*L_ + l] = (unsigned)pos;
  ((float*)ws)[WS_AMVAL + (size_t)b*L_ + l] = amv;
}

/* ---------- pass 3: one wave32 per anchor gathers all outputs (b128 + NT stores) ---------- */
__global__ __launch_bounds__(TPB)
void gather_kernel(const int* __restrict__ gt_labels, const float* __restrict__ gt_bboxes,
                   const float* __restrict__ gt_poses, const float* __restrict__ gt_vertices,
                   const float* __restrict__ gt_rotations, const int* __restrict__ bg_index,
                   const unsigned* __restrict__ ws, float* __restrict__ out)
{
  const size_t a = (size_t)blockIdx.x*(TPB/32) + (threadIdx.x>>5);
  const int lane = threadIdx.x & 31;
  if (a >= (size_t)B_*L_) return;
  const int b = (int)(a / L_);
  const unsigned n   = ws[WS_GIDX + a];
  const unsigned pos = ws[WS_POS  + a];
  const size_t src = (size_t)b*N_ + n;

  constexpr int NV4 = K3_/4;                                /* 51 x 16B per row */
  const v4f* ps = (const v4f*)(gt_poses    + src*K3_);
  const v4f* vs = (const v4f*)(gt_vertices + src*K3_);
  v4f* op = (v4f*)(out + O_POSE + a*K3_);
  v4f* ov = (v4f*)(out + O_VERT + a*K3_);
  if (lane == 0){ __builtin_prefetch(ps, 0, 1); __builtin_prefetch(vs, 0, 1); }
  for (int i=lane; i<NV4; i+=32){ v4f t = ps[i]; __builtin_nontemporal_store(t, op+i); }
  for (int i=lane; i<NV4; i+=32){ v4f t = vs[i]; __builtin_nontemporal_store(t, ov+i); }
  if (lane < 9) __builtin_nontemporal_store(gt_rotations[src*9 + lane], out + O_ROT  + a*9 + lane);
  if (lane < 4) __builtin_nontemporal_store(gt_bboxes[src*4 + lane],    out + O_BBOX + a*4 + lane);
  if (lane == 0){
    const int bg  = bg_index[0];
    const int lbl = pos ? gt_labels[src] : bg;
    float scale = 0.f;
    if (pos){
      float amv = ((const float*)ws)[WS_AMVAL + a];
      float mam = __uint_as_float(ws[WS_MAXAM  + (size_t)b*N_ + n]);
      float mio = __uint_as_float(ws[WS_MAXIOU + (size_t)b*N_ + n]);
      scale = amv/(mam+EPS_)*mio;
    }
    __builtin_nontemporal_store((float)lbl, out + O_LAB + a);
    __builtin_nontemporal_store((lbl != bg && lbl == 0) ? scale : 0.f, out + O_SCORE + a);
    __builtin_nontemporal_store((float)n, out + O_GID + a);
  }
}

extern "C" void kernel_launch(void* const* d_in, const int* in_sizes, int n_in,
                              void* d_out, int out_size, void* d_ws, size_t ws_size,
                              hipStream_t stream) {
  const float* pred_scores   = (const float*)d_in[0];
  const float* pred_bboxes   = (const float*)d_in[1];
  const float* anchor_points = (const float*)d_in[2];
  const int*   gt_labels     = (const int*)  d_in[3];
  const float* gt_bboxes     = (const float*)d_in[4];
  const float* gt_poses      = (const float*)d_in[5];
  const float* gt_vertices   = (const float*)d_in[6];
  const float* gt_rotations  = (const float*)d_in[7];
  const float* pad_gt_mask   = (const float*)d_in[8];
  const int*   bg_index      = (const int*)  d_in[9];
  float*    out = (float*)d_out;
  unsigned* ws  = (unsigned*)d_ws;
  (void)in_sizes; (void)n_in; (void)out_size; (void)ws_size;

  const size_t nvec = WS_ZERO_WORDS/4;
  init_kernel<<<(unsigned)((nvec + 255)/256), 256, 0, stream>>>((v4u*)ws, nvec);

  dim3 g1(N_, B_);
  topk_kernel<<<g1, TPB, 0, stream>>>(pred_scores, pred_bboxes, anchor_points,
                                      gt_bboxes, pad_gt_mask, ws);

  dim3 g2((L_ + TPB - 1)/TPB, B_);
  assign_kernel<<<g2, TPB, 0, stream>>>(pred_scores, pred_bboxes, anchor_points,
                                        gt_bboxes, pad_gt_mask, ws);

  const size_t na = (size_t)B_*L_;
  gather_kernel<<<(unsigned)((na + (TPB/32) - 1)/(TPB/32)), TPB, 0, stream>>>(
      gt_labels, gt_bboxes, gt_poses, gt_vertices, gt_rotations, bg_index, ws, out);
}